// HierBertLayer_2336462209377
// MI455X (gfx1250) — compile-verified
//
#include <hip/hip_runtime.h>
#include <math.h>

// ---------------------------------------------------------------------------
// HierBertLayer for MI455X (gfx1250, wave32, WMMA).
// - All matmuls via V_WMMA_F32_16X16X4_F32 (fp32 matrix pipe; workload is
//   HBM-bandwidth-bound so fp32 WMMA saturates the roofline while matching
//   fp32 reference numerics).
// - A/W tiles staged global->LDS with GLOBAL_LOAD_ASYNC_TO_LDS_B128 under a
//   double-buffered pipeline: exactly 13 async issues per wave per chunk, so
//   `s_wait_asynccnt 13` retires chunk i while chunk i+1 is in flight.
// - V is produced transposed (VT[b][dim][key]) so the P@V B-fragments are
//   single contiguous b64 loads.
// ---------------------------------------------------------------------------

typedef __attribute__((ext_vector_type(2))) float v2f;
typedef __attribute__((ext_vector_type(8))) float v8f;

#define WMMA_F32(a, b, c) \
  __builtin_amdgcn_wmma_f32_16x16x4_f32(false, (a), false, (b), (short)0, (c), false, false)

// async 16B global -> LDS copy (per-lane), tracked by ASYNCcnt
__device__ __forceinline__ void async_g2l_b128(void* lds, const void* gaddr) {
  unsigned ldst = (unsigned)(uintptr_t)lds;  // low 32b = LDS offset
  unsigned long long ga = (unsigned long long)(uintptr_t)gaddr;
  asm volatile("global_load_async_to_lds_b128 %0, %1, off" ::"v"(ldst), "v"(ga)
               : "memory");
}
__device__ __forceinline__ void wait_async_le13() {
  asm volatile("s_wait_asynccnt 0xd" ::: "memory");
}
__device__ __forceinline__ void wait_async0() {
  asm volatile("s_wait_asynccnt 0x0" ::: "memory");
}

static constexpr int Hs  = 768;
static constexpr int FFs = 3072;
static constexpr int Ss  = 512;
static constexpr int KC  = 16;    // K chunk staged per pipeline stage
static constexpr int WPAD = 784;  // LDS row stride for W chunk (bank-split halves)
static constexpr int APAD = 20;   // LDS row stride for A chunk (conflict-free)

// ---------------------------------------------------------------------------
// Fused GEMM: 16-row strip per workgroup, 8 waves x 96 columns (6 tiles).
// EPI 0: out = A@W + bias
// EPI 1: out = gelu(A@W + bias)
// EPI 2: out = LN(A@W + bias + res (+extra)) [* (hm!=0)]   (N must be 768)
// EPI 3: out^T = A@W + bias   (store [N][M] as VT[b][dim][key], M=1024,S=512)
// ---------------------------------------------------------------------------
template <int EPI>
__global__ __launch_bounds__(256) void gemm16_kernel(
    const float* __restrict__ A, const float* __restrict__ W,
    const float* __restrict__ bias, const float* __restrict__ res,
    const float* __restrict__ extra, const float* __restrict__ gamma,
    const float* __restrict__ beta, const int* __restrict__ hm,
    float* __restrict__ out, int M, int K, int N) {
  __shared__ __align__(16) float Wl[2][16 * WPAD];   // ~98KB double buffer
  __shared__ __align__(16) float Alds[2][16 * APAD];
  __shared__ float rsum[(EPI == 2) ? 256 : 1];
  __shared__ float rsq [(EPI == 2) ? 256 : 1];
  __shared__ float rmean[(EPI == 2) ? 16 : 1];
  __shared__ float rrstd[(EPI == 2) ? 16 : 1];
  float* Slds = &Wl[0][0];  // EPI2 LN staging aliases dead W buf0 (16*768 <= 16*WPAD)

  const int tid  = threadIdx.x;
  const int wave = tid >> 5;
  const int lane = tid & 31;
  const int ln   = lane & 15;
  const int hi   = lane >> 4;
  const int m0   = blockIdx.x * 16;

  for (int nc0 = 0; nc0 < N; nc0 += 768) {
    // stage chunk (kc) into LDS buffer bsel: exactly 13 async issues per wave
    auto stage = [&](int kc, int bsel) {
      {  // A tile: 64 x b128, duplicated 4x so every wave issues 1 (full EXEC)
        const int blk = tid & 63;
        const int r = blk >> 2, c4 = (blk & 3) * 4;
        async_g2l_b128(&Alds[bsel][r * APAD + c4],
                       A + (size_t)(m0 + r) * K + kc + c4);
      }
      // W chunk 16x768: 3072 x b128 => 12 issues per wave
      for (int i = tid; i < 16 * 192; i += 256) {
        const int r = i / 192, c4 = (i % 192) * 4;
        async_g2l_b128(&Wl[bsel][r * WPAD + c4],
                       W + (size_t)(kc + r) * N + nc0 + c4);
      }
    };

    v8f acc[6];
#pragma unroll
    for (int t = 0; t < 6; ++t)
#pragma unroll
      for (int j = 0; j < 8; ++j) acc[t][j] = 0.0f;

    const int nch = K / KC;
    stage(0, 0);  // prologue (buffers free: trailing barrier of previous chunk)
    for (int ic = 0; ic < nch; ++ic) {
      const int cur = ic & 1;
      if (ic + 1 < nch) {
        stage((ic + 1) * KC, cur ^ 1);
        wait_async_le13();  // chunk ic landed (in-order completion)
      } else {
        wait_async0();
      }
      __syncthreads();  // all waves' copies for chunk ic visible
#pragma unroll
      for (int kk = 0; kk < KC; kk += 4) {
        const int k0 = kk + 2 * hi;
        const v2f a = *(const v2f*)&Alds[cur][ln * APAD + k0];
        v2f bb[6];
#pragma unroll
        for (int t = 0; t < 6; ++t) {
          const int cl = wave * 96 + t * 16 + ln;
          bb[t].x = Wl[cur][k0 * WPAD + cl];
          bb[t].y = Wl[cur][(k0 + 1) * WPAD + cl];
        }
#pragma unroll
        for (int t = 0; t < 6; ++t) acc[t] = WMMA_F32(a, bb[t], acc[t]);
      }
      __syncthreads();  // chunk ic reads done before its buffer is re-staged
    }

    if (EPI == 0 || EPI == 1) {
#pragma unroll
      for (int t = 0; t < 6; ++t) {
        const int colg = nc0 + wave * 96 + t * 16 + ln;
#pragma unroll
        for (int j = 0; j < 8; ++j) {
          const int row = j + 8 * hi;
          float v = acc[t][j] + bias[colg];
          if (EPI == 1) v = 0.5f * v * (1.0f + erff(v * 0.70710678118654752f));
          out[(size_t)(m0 + row) * N + colg] = v;
        }
      }
    } else if (EPI == 3) {  // transposed store: VT[b][dim][key]
#pragma unroll
      for (int t = 0; t < 6; ++t) {
        const int colg = nc0 + wave * 96 + t * 16 + ln;
#pragma unroll
        for (int j = 0; j < 8; ++j) {
          const int row = j + 8 * hi;
          const int m = m0 + row;
          out[((size_t)(m >> 9) * Hs + colg) * Ss + (m & (Ss - 1))] =
              acc[t][j] + bias[colg];
        }
      }
    } else {  // EPI == 2 : residual + LayerNorm over the 768-wide row
#pragma unroll
      for (int t = 0; t < 6; ++t) {
        const int col = wave * 96 + t * 16 + ln;  // nc0 == 0 here (N==768)
#pragma unroll
        for (int j = 0; j < 8; ++j) {
          const int row = j + 8 * hi;
          float v = acc[t][j] + bias[col] + res[(size_t)(m0 + row) * Hs + col];
          if (extra) v += extra[(size_t)(m0 + row) * Hs + col];
          Slds[row * Hs + col] = v;
        }
      }
      __syncthreads();
      {
        const int row = tid >> 4, sub = tid & 15;
        float s = 0.f, q = 0.f;
        for (int c = sub; c < Hs; c += 16) {
          const float x = Slds[row * Hs + c];
          s += x;
          q += x * x;
        }
        rsum[row * 16 + sub] = s;
        rsq[row * 16 + sub]  = q;
      }
      __syncthreads();
      if (tid < 16) {
        float s = 0.f, q = 0.f;
        for (int i = 0; i < 16; ++i) { s += rsum[tid * 16 + i]; q += rsq[tid * 16 + i]; }
        const float mu  = s * (1.0f / Hs);
        const float var = q * (1.0f / Hs) - mu * mu;
        rmean[tid] = mu;
        rrstd[tid] = rsqrtf(var + 1e-12f);
      }
      __syncthreads();
      {
        const int row = tid >> 4, sub = tid & 15;
        const float mu = rmean[row], rs = rrstd[row];
        float mult = 1.0f;
        if (hm) mult = (hm[m0 + row] != 0) ? 1.0f : 0.0f;
        for (int c = sub; c < Hs; c += 16) {
          const float v = (Slds[row * Hs + c] - mu) * rs * gamma[c] + beta[c];
          out[(size_t)(m0 + row) * Hs + c] = v * mult;
        }
      }
    }
  }
}

// ---------------------------------------------------------------------------
// Attention: one workgroup per (16-query tile, head, batch).
// Scores via WMMA into LDS (16x512), softmax, then P@V via WMMA with V
// consumed from the transposed VT[b][dim][key] buffer (contiguous b64 frags).
// HIER=true : per-(q,k) bias from hier_mask equality (compacted hier path).
// HIER=false: per-key bias from attention_mask.
// ---------------------------------------------------------------------------
template <bool HIER>
__global__ __launch_bounds__(256) void attn_kernel(
    const float* __restrict__ Q, const float* __restrict__ K,
    const float* __restrict__ VT, const float* __restrict__ amask,
    const int* __restrict__ hmask, float* __restrict__ ctx) {
  __shared__ __align__(16) float Sl[16][Ss];
  __shared__ float red[16][16];
  __shared__ float rowmax[16], rowsum[16];
  __shared__ int qgm[16];

  const int tid = threadIdx.x, wave = tid >> 5, lane = tid & 31;
  const int ln = lane & 15, hi = lane >> 4;
  const int q0 = blockIdx.x * 16, h = blockIdx.y, b = blockIdx.z;
  const size_t base = (size_t)b * Ss * Hs;
  const int hoff = h * 64;

  if (HIER && tid < 16) qgm[tid] = hmask[b * Ss + q0 + tid];
  __syncthreads();

  // ---- stage 1: scores = Q K^T / 8 + bias ; each wave: 64 keys (4 tiles)
  v2f aq[16];
#pragma unroll
  for (int k16 = 0; k16 < 16; ++k16)
    aq[k16] = *(const v2f*)(Q + base + (size_t)(q0 + ln) * Hs + hoff + k16 * 4 + 2 * hi);

  v8f acc[4];
#pragma unroll
  for (int t = 0; t < 4; ++t)
#pragma unroll
    for (int j = 0; j < 8; ++j) acc[t][j] = 0.0f;

#pragma unroll
  for (int k16 = 0; k16 < 16; ++k16) {
#pragma unroll
    for (int t = 0; t < 4; ++t) {
      const int key = wave * 64 + t * 16 + ln;
      const v2f bf = *(const v2f*)(K + base + (size_t)key * Hs + hoff + k16 * 4 + 2 * hi);
      acc[t] = WMMA_F32(aq[k16], bf, acc[t]);
    }
  }
#pragma unroll
  for (int t = 0; t < 4; ++t) {
    const int key = wave * 64 + t * 16 + ln;
    int kg = 0;
    float kb = 0.0f;
    if (HIER) kg = hmask[b * Ss + key];
    else      kb = (1.0f - amask[b * Ss + key]) * -10000.0f;
#pragma unroll
    for (int j = 0; j < 8; ++j) {
      const int row = j + 8 * hi;
      const float biasv = HIER ? ((kg == qgm[row]) ? 0.0f : -10000.0f) : kb;
      Sl[row][key] = acc[t][j] * 0.125f + biasv;
    }
  }
  __syncthreads();

  // ---- stage 2: softmax over 512 keys per row
  {
    const int row = tid >> 4, sub = tid & 15;
    float m = -3.0e38f;
    for (int c = sub; c < Ss; c += 16) m = fmaxf(m, Sl[row][c]);
    red[row][sub] = m;
  }
  __syncthreads();
  if (tid < 16) {
    float m = -3.0e38f;
    for (int i = 0; i < 16; ++i) m = fmaxf(m, red[tid][i]);
    rowmax[tid] = m;
  }
  __syncthreads();
  {
    const int row = tid >> 4, sub = tid & 15;
    const float m = rowmax[row];
    float s = 0.0f;
    for (int c = sub; c < Ss; c += 16) {
      const float e = expf(Sl[row][c] - m);
      Sl[row][c] = e;
      s += e;
    }
    red[row][sub] = s;
  }
  __syncthreads();
  if (tid < 16) {
    float s = 0.0f;
    for (int i = 0; i < 16; ++i) s += red[tid][i];
    rowsum[tid] = s;
  }
  __syncthreads();

  // ---- stage 3: ctx = P V ; waves 0..3 each own 16 head dims
  if (wave < 4) {
    const float* vtb = VT + ((size_t)b * Hs + hoff + wave * 16 + ln) * Ss;
    v8f c2;
#pragma unroll
    for (int j = 0; j < 8; ++j) c2[j] = 0.0f;
    for (int kk = 0; kk < Ss; kk += 4) {
      const v2f a  = *(const v2f*)&Sl[ln][kk + 2 * hi];
      const v2f bf = *(const v2f*)&vtb[kk + 2 * hi];  // contiguous (key-major)
      c2 = WMMA_F32(a, bf, c2);
    }
#pragma unroll
    for (int j = 0; j < 8; ++j) {
      const int row = j + 8 * hi;
      ctx[base + (size_t)(q0 + row) * Hs + hoff + wave * 16 + ln] = c2[j] / rowsum[row];
    }
  }
}

// ---------------------------------------------------------------------------
extern "C" void kernel_launch(void* const* d_in, const int* in_sizes, int n_in,
                              void* d_out, int out_size, void* d_ws, size_t ws_size,
                              hipStream_t stream) {
  (void)in_sizes; (void)n_in; (void)out_size; (void)ws_size;
  const float* x     = (const float*)d_in[0];
  const float* amask = (const float*)d_in[1];
  const float* qw   = (const float*)d_in[2];  const float* qb   = (const float*)d_in[3];
  const float* kw   = (const float*)d_in[4];  const float* kb   = (const float*)d_in[5];
  const float* vw   = (const float*)d_in[6];  const float* vb   = (const float*)d_in[7];
  const float* aow  = (const float*)d_in[8];  const float* aob  = (const float*)d_in[9];
  const float* alng = (const float*)d_in[10]; const float* alnb = (const float*)d_in[11];
  const float* iw   = (const float*)d_in[12]; const float* ib   = (const float*)d_in[13];
  const float* ow   = (const float*)d_in[14]; const float* ob   = (const float*)d_in[15];
  const float* olng = (const float*)d_in[16]; const float* olnb = (const float*)d_in[17];
  const float* hqw   = (const float*)d_in[18]; const float* hqb   = (const float*)d_in[19];
  const float* hkw   = (const float*)d_in[20]; const float* hkb   = (const float*)d_in[21];
  const float* hvw   = (const float*)d_in[22]; const float* hvb   = (const float*)d_in[23];
  const float* haow  = (const float*)d_in[24]; const float* haob  = (const float*)d_in[25];
  const float* halng = (const float*)d_in[26]; const float* halnb = (const float*)d_in[27];
  const float* hiw   = (const float*)d_in[28]; const float* hib   = (const float*)d_in[29];
  const float* how   = (const float*)d_in[30]; const float* hob   = (const float*)d_in[31];
  const float* holng = (const float*)d_in[32]; const float* holnb = (const float*)d_in[33];
  const int*   hmask = (const int*)d_in[34];
  float* out = (float*)d_out;

  // workspace (floats): Q,K,VT,CTX,A,HIER_OUT each 1024*768; H 1024*3072
  float* ws = (float*)d_ws;
  const size_t U = (size_t)1024 * Hs;
  float* Qb = ws;         float* Kb = ws + U;     float* VTb = ws + 2 * U;
  float* CT = ws + 3 * U; float* Ab = ws + 4 * U; float* HO  = ws + 5 * U;
  float* Hb = ws + 6 * U;  // 1024*3072

  const dim3 blk(256);
  const dim3 gstrip(64);        // 1024 rows / 16
  const dim3 gattn(32, 12, 2);  // q-tiles, heads, batch
  const int M = 1024;

  // ---- hier path (compacted: one batch-worth, per-query group key mask) ----
  gemm16_kernel<0><<<gstrip, blk, 0, stream>>>(x, hqw, hqb, nullptr, nullptr, nullptr, nullptr, nullptr, Qb, M, Hs, Hs);
  gemm16_kernel<0><<<gstrip, blk, 0, stream>>>(x, hkw, hkb, nullptr, nullptr, nullptr, nullptr, nullptr, Kb, M, Hs, Hs);
  gemm16_kernel<3><<<gstrip, blk, 0, stream>>>(x, hvw, hvb, nullptr, nullptr, nullptr, nullptr, nullptr, VTb, M, Hs, Hs);
  attn_kernel<true><<<gattn, blk, 0, stream>>>(Qb, Kb, VTb, nullptr, hmask, CT);
  gemm16_kernel<2><<<gstrip, blk, 0, stream>>>(CT, haow, haob, x, nullptr, halng, halnb, nullptr, Ab, M, Hs, Hs);
  gemm16_kernel<1><<<gstrip, blk, 0, stream>>>(Ab, hiw, hib, nullptr, nullptr, nullptr, nullptr, nullptr, Hb, M, Hs, FFs);
  gemm16_kernel<2><<<gstrip, blk, 0, stream>>>(Hb, how, hob, Ab, nullptr, holng, holnb, hmask, HO, M, FFs, Hs);

  // ---- main path (double residual adds HO before final LN) ----
  gemm16_kernel<0><<<gstrip, blk, 0, stream>>>(x, qw, qb, nullptr, nullptr, nullptr, nullptr, nullptr, Qb, M, Hs, Hs);
  gemm16_kernel<0><<<gstrip, blk, 0, stream>>>(x, kw, kb, nullptr, nullptr, nullptr, nullptr, nullptr, Kb, M, Hs, Hs);
  gemm16_kernel<3><<<gstrip, blk, 0, stream>>>(x, vw, vb, nullptr, nullptr, nullptr, nullptr, nullptr, VTb, M, Hs, Hs);
  attn_kernel<false><<<gattn, blk, 0, stream>>>(Qb, Kb, VTb, amask, nullptr, CT);
  gemm16_kernel<2><<<gstrip, blk, 0, stream>>>(CT, aow, aob, x, nullptr, alng, alnb, nullptr, Ab, M, Hs, Hs);
  gemm16_kernel<1><<<gstrip, blk, 0, stream>>>(Ab, iw, ib, nullptr, nullptr, nullptr, nullptr, nullptr, Hb, M, Hs, FFs);
  gemm16_kernel<2><<<gstrip, blk, 0, stream>>>(Hb, ow, ob, Ab, HO, olng, olnb, nullptr, out, M, FFs, Hs);
}